// DWTFrontEnd_63127429316910
// MI455X (gfx1250) — compile-verified
//
#include <hip/hip_runtime.h>
#include <stdint.h>

// ---------------------------------------------------------------------------
// 3-level db4 DWT front end, fused single kernel for MI455X (gfx1250).
//
// Bandwidth-bound: 64 MB in + 256 MB out => ~14 us floor at 23.3 TB/s.
// Strategy: one workgroup per (b,c) row; stage the 32 KB row into LDS with
// CDNA5 async global->LDS loads (ASYNCcnt), run the whole analysis +
// 4 reconstructions inside LDS (peak 48 KB, carefully re-used), and stream
// each output element exactly once with non-temporal stores so the 256 MB
// output doesn't evict the working set from the 192 MB L2.
// ---------------------------------------------------------------------------

#define TPB 256

// Fixed geometry from the reference (x = (16,128,8192) f32, LEVELS=3, K=8).
#define L0v 8192
#define L1v 4100   // L0/2 + 4
#define L2v 2054   // L1/2 + 4
#define L3v 1031   // L2/2 + 4

// LDS layout (float indices). Peak live set = 12292 floats = 48.02 KiB.
//   OX : x row            [0,    8192)   dead once a1 is computed
//   OP : d1 -> a1 -> t4100[8192, 12292)
//   OQ : d2 -> t2054      [0,    2054)   (reuses dead x region)
//   ORr: a2               [2054, 4108)
//   OD3: d3               [4108, 5139)
//   OA3: a3               [5200, 6231)
#define OX   0
#define OP   8192
#define OQ   0
#define ORr  2054
#define OD3  4108
#define OA3  5200
#define SMEMF 12292

typedef __attribute__((ext_vector_type(2))) float f32x2;
typedef __attribute__((ext_vector_type(4))) int   v4i_t;

__device__ constexpr float DLO[8] = {-0.0105974018f,  0.0328830117f,  0.0308413818f, -0.1870348117f,
                                     -0.0279837694f,  0.6308807679f,  0.7148465706f,  0.2303778133f};
__device__ constexpr float DHI[8] = {-0.2303778133f,  0.7148465706f, -0.6308807679f, -0.0279837694f,
                                      0.1870348117f,  0.0308413818f, -0.0328830117f, -0.0105974018f};
__device__ constexpr float RLO[8] = { 0.2303778133f,  0.7148465706f,  0.6308807679f, -0.0279837694f,
                                     -0.1870348117f,  0.0308413818f,  0.0328830117f, -0.0105974018f};
__device__ constexpr float RHI[8] = {-0.0105974018f, -0.0328830117f,  0.0308413818f,  0.1870348117f,
                                     -0.0279837694f, -0.6308807679f,  0.7148465706f, -0.2303778133f};

// ---- CDNA5 async global->LDS staging (ASYNCcnt path) ----------------------
__device__ __forceinline__ void async_ld_b128(void* lds_dst, const void* gsrc) {
#if __has_builtin(__builtin_amdgcn_global_load_async_to_lds_b128)
  __builtin_amdgcn_global_load_async_to_lds_b128(
      (__attribute__((address_space(1))) v4i_t*)(uintptr_t)gsrc,
      (__attribute__((address_space(3))) v4i_t*)(uintptr_t)lds_dst,
      0, 0);
#else
  unsigned l = (unsigned)(uintptr_t)lds_dst;  // low 32 bits of generic LDS addr = LDS byte offset
  asm volatile("global_load_async_to_lds_b128 %0, %1, off" :: "v"(l), "v"(gsrc) : "memory");
#endif
}

__device__ __forceinline__ void wait_async0() {
#if __has_builtin(__builtin_amdgcn_s_wait_asynccnt)
  __builtin_amdgcn_s_wait_asynccnt(0);
#else
  asm volatile("s_wait_asynccnt 0x0" ::: "memory");
#endif
}

// ---- forward analysis: y[i] = sum_k h[k] * x[refl(2i-7+k)] ----------------
__device__ __forceinline__ int refl(int t, int L) {
  t = (t < 0) ? -t : t;
  t = (t >= L) ? (2 * L - 2 - t) : t;
  return t;
}

__device__ __forceinline__ void analyze1(const float* __restrict__ s, float* __restrict__ d,
                                         const float* h, int L, int Lout) {
  for (int i = threadIdx.x; i < Lout; i += TPB) {
    const int base = 2 * i - 7;
    float acc = 0.f;
#pragma unroll
    for (int k = 0; k < 8; ++k)
      acc = fmaf(h[k], s[refl(base + k, L)], acc);
    d[i] = acc;
  }
}

__device__ __forceinline__ void analyze2(const float* __restrict__ s,
                                         float* __restrict__ dlo, float* __restrict__ dhi,
                                         int L, int Lout) {
  for (int i = threadIdx.x; i < Lout; i += TPB) {
    const int base = 2 * i - 7;
    float lo = 0.f, hi = 0.f;
#pragma unroll
    for (int k = 0; k < 8; ++k) {
      const float v = s[refl(base + k, L)];
      lo = fmaf(DLO[k], v, lo);
      hi = fmaf(DHI[k], v, hi);
    }
    dlo[i] = lo;
    dhi[i] = hi;
  }
}

// ---- polyphase upsample (conv_transpose stride 2 + center crop of 7) ------
// T = 2*Lc - 8;  rec[2t]   = w7 c[t]   + w5 c[t+1] + w3 c[t+2] + w1 c[t+3]
//                rec[2t+1] = w6 c[t+1] + w4 c[t+2] + w2 c[t+3] + w0 c[t+4]
template <bool HI>
__device__ __forceinline__ void up_to_lds(const float* __restrict__ c, float* __restrict__ d, int Lc) {
  const float* W = HI ? RHI : RLO;
  for (int t = threadIdx.x; t < Lc - 4; t += TPB) {
    const float c0 = c[t], c1 = c[t + 1], c2 = c[t + 2], c3 = c[t + 3], c4 = c[t + 4];
    d[2 * t]     = fmaf(W[7], c0, fmaf(W[5], c1, fmaf(W[3], c2, W[1] * c3)));
    d[2 * t + 1] = fmaf(W[6], c1, fmaf(W[4], c2, fmaf(W[2], c3, W[0] * c4)));
  }
}

template <bool HI>
__device__ __forceinline__ void up_to_glb(const float* __restrict__ c, float* __restrict__ o, int Lc) {
  const float* W = HI ? RHI : RLO;
  for (int t = threadIdx.x; t < Lc - 4; t += TPB) {
    const float c0 = c[t], c1 = c[t + 1], c2 = c[t + 2], c3 = c[t + 3], c4 = c[t + 4];
    f32x2 v;
    v.x = fmaf(W[7], c0, fmaf(W[5], c1, fmaf(W[3], c2, W[1] * c3)));
    v.y = fmaf(W[6], c1, fmaf(W[4], c2, fmaf(W[2], c3, W[0] * c4)));
    __builtin_nontemporal_store(v, (f32x2*)(o + 2 * t));  // stream past L2
  }
}

// ---------------------------------------------------------------------------
__global__ __launch_bounds__(TPB) void dwt_db4_bands(const float* __restrict__ x,
                                                     float* __restrict__ out,
                                                     int rows) {
  __shared__ float sm[SMEMF];
  const int row = blockIdx.x;
  const float* xr = x + (size_t)row * L0v;
  const size_t band = (size_t)rows * L0v;
  float* o0 = out + 0 * band + (size_t)row * L0v;  // approx band
  float* o1 = out + 1 * band + (size_t)row * L0v;  // detail level 3
  float* o2 = out + 2 * band + (size_t)row * L0v;  // detail level 2
  float* o3 = out + 3 * band + (size_t)row * L0v;  // detail level 1

  // 1) stage x row (32 KB) into LDS via async global->LDS loads, 16 B/lane.
  for (int c = threadIdx.x; c < L0v / 4; c += TPB)
    async_ld_b128(&sm[OX + 4 * c], xr + 4 * c);
  wait_async0();
  __syncthreads();

  // 2) d1 = DWT_hi(x)
  analyze1(&sm[OX], &sm[OP], DHI, L0v, L1v);
  __syncthreads();

  // 3) band 3 = crop(up_hi(d1)) -> out   (then OP is free again)
  up_to_glb<true>(&sm[OP], o3, L1v);
  __syncthreads();

  // 4) a1 = DWT_lo(x)   (overwrites d1; x dead after this)
  analyze1(&sm[OX], &sm[OP], DLO, L0v, L1v);
  __syncthreads();

  // 5) {a2, d2} = DWT(a1)   (written into the dead x region)
  analyze2(&sm[OP], &sm[ORr], &sm[OQ], L1v, L2v);
  __syncthreads();

  // 6) band 2: up_hi(d2) -> t4100(OP), up_lo -> out
  up_to_lds<true>(&sm[OQ], &sm[OP], L2v);
  __syncthreads();
  up_to_glb<false>(&sm[OP], o2, L1v);
  __syncthreads();

  // 7) {a3, d3} = DWT(a2)
  analyze2(&sm[ORr], &sm[OA3], &sm[OD3], L2v, L3v);
  __syncthreads();

  // 8) band 1: up_hi(d3) -> t2054(OQ) -> up_lo -> t4100(OP) -> up_lo -> out
  up_to_lds<true>(&sm[OD3], &sm[OQ], L3v);
  __syncthreads();
  up_to_lds<false>(&sm[OQ], &sm[OP], L2v);
  __syncthreads();
  up_to_glb<false>(&sm[OP], o1, L1v);
  __syncthreads();

  // 9) band 0: up_lo(a3) -> t2054(OQ) -> t4100(OP) -> out
  up_to_lds<false>(&sm[OA3], &sm[OQ], L3v);
  __syncthreads();
  up_to_lds<false>(&sm[OQ], &sm[OP], L2v);
  __syncthreads();
  up_to_glb<false>(&sm[OP], o0, L1v);
}

extern "C" void kernel_launch(void* const* d_in, const int* in_sizes, int n_in,
                              void* d_out, int out_size, void* d_ws, size_t ws_size,
                              hipStream_t stream) {
  (void)n_in; (void)out_size; (void)d_ws; (void)ws_size;
  const float* x = (const float*)d_in[0];
  float* out = (float*)d_out;
  const int rows = in_sizes[0] / L0v;  // 16 * 128 = 2048
  dwt_db4_bands<<<rows, TPB, 0, stream>>>(x, out, rows);
}